// CatFeaturesItemNet_47459388620976
// MI455X (gfx1250) — compile-verified
//
#include <hip/hip_runtime.h>
#include <hip/hip_bf16.h>
#include <stdint.h>

// ---------------------------------------------------------------------------
// EmbeddingBag(sum) via CDNA5 Tensor Data Mover gather-mode.
//   out[b, :] = sum_{k < len(items[b])} weight[ emb_bag_inputs[off(items[b]) + k], : ]
// Gather of up to 16 rows x 512B per item is done by TENSOR_LOAD_TO_LDS with
// gather_mode=1 / 32-bit indices (2 descriptors x 8 rows), double-buffered in
// LDS per wave and synchronized with s_wait_tensorcnt. Output rows are stored
// non-temporally so the 51MB weight table stays resident in the 192MB L2.
// ---------------------------------------------------------------------------

#define L_MAX     16
#define DFACT     128                 // N_FACTORS
#define ROW_BYTES (DFACT * 4)         // 512 B per table row
#define NVALS     100000              // n_cat_feature_values (OOB bound only)
#define WPB       4                   // waves per block
#define THREADS   (WPB * 32)
#define IPW       4                   // items processed per wave

typedef unsigned int u32;
typedef u32   __attribute__((ext_vector_type(4))) v4u;
typedef int   __attribute__((ext_vector_type(4))) v4i;
typedef int   __attribute__((ext_vector_type(8))) v8i;
typedef float __attribute__((ext_vector_type(4))) v4f;   // native vector (builtin-legal)

static __device__ __forceinline__ void tdm_issue(v4u g0, v8i g1, v4i g2, v4i g3) {
#if defined(__clang_major__) && (__clang_major__ >= 23)
  v8i g4 = {0, 0, 0, 0, 0, 0, 0, 0};
  __builtin_amdgcn_tensor_load_to_lds(g0, g1, g2, g3, g4, 0);
#else
  __builtin_amdgcn_tensor_load_to_lds(g0, g1, g2, g3, 0);
#endif
}

// One gather descriptor: up to 8 rows (32-bit row indices) of 128 fp32 each,
// from `wbase` table into LDS at byte offset `ldsoff`.
static __device__ __forceinline__ void issue_gather8(const int* ids, int nvalid,
                                                     uint64_t wbase, u32 ldsoff) {
  v4u g0;
  g0.x = 0xC0000001u;                                       // count=1, gather_mode=1, 32b idx
  g0.y = ldsoff;                                            // lds_addr (bytes)
  g0.z = (u32)(wbase & 0xFFFFFFFFull);                      // global_addr[31:0]
  g0.w = (u32)((wbase >> 32) & 0x1FFFFFFull) | (2u << 30);  // global_addr[56:32] | type=2

  v8i g1;
  g1[0] = (int)(2u << 16);                                  // data_size = 4 B
  g1[1] = (int)((u32)DFACT << 16);                          // tensor_dim0 = 128 (lo16 in [31:16])
  g1[2] = (int)(((u32)NVALS & 0xFFFFu) << 16);              // tensor_dim0 hi=0 | tensor_dim1 lo
  g1[3] = (int)(((u32)NVALS >> 16) | ((u32)DFACT << 16));   // tensor_dim1 hi | tile_dim0 = 128
  g1[4] = nvalid & 0xFFFF;                                  // tile_dim1 = #valid indices
  g1[5] = DFACT;                                            // tensor_dim0_stride = 128 elems
  g1[6] = 0;
  g1[7] = 0;

  v4i g2 = { ids[0], ids[1], ids[2], ids[3] };              // row_index_0..3 (32-bit mode)
  v4i g3 = { ids[4], ids[5], ids[6], ids[7] };              // row_index_4..7
  tdm_issue(g0, g1, g2, g3);
}

// Read the (uniform) metadata for batch slot b, issue both gather descriptors,
// return the item's length. All addresses are wave-uniform, so the compiler
// turns the id fetches into s_load (SMEM) and builds the D# in SALU.
static __device__ __forceinline__ int issue_item(
    int b, const int* __restrict__ items, const int* __restrict__ offsets,
    const int* __restrict__ lens, const int* __restrict__ ebi,
    uint64_t wbase, u32 ldsoff) {
  int it  = __builtin_amdgcn_readfirstlane(items[b]);
  int off = __builtin_amdgcn_readfirstlane(offsets[it]);
  int len = __builtin_amdgcn_readfirstlane(lens[it]);
  int ids[L_MAX];
#pragma unroll
  for (int k = 0; k < L_MAX; ++k)      // slots [16*i, 16*i+16) always in-bounds
    ids[k] = __builtin_amdgcn_readfirstlane(ebi[off + k]);
  int n0 = (len > 8) ? 8 : len;
  int n1 = (len > 8) ? (len - 8) : 0;
  issue_gather8(&ids[0], n0, wbase, ldsoff);
  issue_gather8(&ids[8], n1, wbase, ldsoff + 8u * ROW_BYTES);
  return len;
}

__global__ __launch_bounds__(THREADS) void embbag_tdm_kernel(
    const int* __restrict__ items, const int* __restrict__ ebi,
    const int* __restrict__ offsets, const int* __restrict__ lens,
    const float* __restrict__ weight, float* __restrict__ out, int batch) {
  __shared__ float lds[WPB * 2 * L_MAX * DFACT];  // 4 waves x 2 bufs x 8KB = 64KB

  const int lane  = threadIdx.x & 31;
  const int wib   = threadIdx.x >> 5;
  const int gwave = blockIdx.x * WPB + wib;
  const int base  = gwave * IPW;
  if (base >= batch) return;

  float* buf0 = &lds[(wib * 2 + 0) * L_MAX * DFACT];
  float* buf1 = &lds[(wib * 2 + 1) * L_MAX * DFACT];
  const u32 off0 = (u32)(uintptr_t)buf0;   // low 32 bits of shared addr = LDS offset
  const u32 off1 = (u32)(uintptr_t)buf1;
  const uint64_t wbase = (uint64_t)(uintptr_t)weight;

  int lenbuf[2];
  // Prologue: start the DMA gather for the first item.
  lenbuf[0] = issue_item(base, items, offsets, lens, ebi, wbase, off0);

#pragma unroll
  for (int j = 0; j < IPW; ++j) {
    int b = base + j;
    if (b >= batch) break;
    const bool has_next = (j + 1 < IPW) && (b + 1 < batch);
    if (has_next) {
      // Kick off item j+1's gathers into the other buffer, then wait until at
      // most the newest pair is outstanding => item j's pair is complete
      // (tensor ops from one wave complete in-order).
      lenbuf[(j + 1) & 1] = issue_item(b + 1, items, offsets, lens, ebi, wbase,
                                       ((j + 1) & 1) ? off1 : off0);
      __builtin_amdgcn_s_wait_tensorcnt(2);
    } else {
      __builtin_amdgcn_s_wait_tensorcnt(0);
    }
    asm volatile("" ::: "memory");  // TDM wrote LDS behind the compiler's back

    const v4f* src = (const v4f*)((j & 1) ? buf1 : buf0);
    const int len = lenbuf[j & 1];               // wave-uniform trip count
    v4f acc = {0.f, 0.f, 0.f, 0.f};
    for (int k = 0; k < len; ++k)                // sum the gathered rows
      acc += src[k * (DFACT / 4) + lane];        // ds_load_b128, conflict-free
    // Write-once output stream: non-temporal store keeps the weight table's
    // gather working set resident in L2 instead of being evicted by output.
    __builtin_nontemporal_store(acc, (v4f*)(out + (size_t)b * DFACT) + lane);
  }
}

extern "C" void kernel_launch(void* const* d_in, const int* in_sizes, int n_in,
                              void* d_out, int out_size, void* d_ws, size_t ws_size,
                              hipStream_t stream) {
  const int*   items   = (const int*)d_in[0];
  const int*   ebi     = (const int*)d_in[1];
  const int*   offsets = (const int*)d_in[2];
  const int*   lens    = (const int*)d_in[3];
  const float* weight  = (const float*)d_in[4];
  float* out = (float*)d_out;

  const int batch  = in_sizes[0];
  const int waves  = (batch + IPW - 1) / IPW;
  const int blocks = (waves + WPB - 1) / WPB;
  embbag_tdm_kernel<<<blocks, THREADS, 0, stream>>>(items, ebi, offsets, lens,
                                                    weight, out, batch);
}